// HarrisNet_57166014710236
// MI455X (gfx1250) — compile-verified
//
#include <hip/hip_runtime.h>
#include <hip/hip_bf16.h>

typedef __attribute__((ext_vector_type(2))) float v2f;
typedef __attribute__((ext_vector_type(8))) float v8f;
typedef __attribute__((ext_vector_type(4))) unsigned u32x4;
typedef __attribute__((ext_vector_type(8))) unsigned u32x8;

// ---------------- Gaussian 7-tap (sigma=5), normalized, compile-time ----------------
constexpr double E1 = 0.98019867330675525;   // exp(-1/50)
constexpr double E2 = 0.92311634638663587;   // exp(-4/50)
constexpr double E3 = 0.83527021141127423;   // exp(-9/50)
constexpr double GS = 1.0 + 2.0 * (E1 + E2 + E3);
__device__ __constant__ float GW[7] = {
    (float)(E3 / GS), (float)(E2 / GS), (float)(E1 / GS), (float)(1.0 / GS),
    (float)(E1 / GS), (float)(E2 / GS), (float)(E3 / GS)};

__device__ __forceinline__ float band(int m, int k) {
  int d = k - m;
  return (d >= 0 && d < 7) ? GW[d] : 0.0f;
}

__device__ __forceinline__ v8f wmma4(v2f a, v2f b, v8f c) {
  // D = A(16x4,f32) * B(4x16,f32) + C(16x16,f32)
  return __builtin_amdgcn_wmma_f32_16x16x4_f32(
      /*neg_a=*/false, a, /*neg_b=*/false, b,
      /*c_mod=*/(short)0, c, /*reuse_a=*/false, /*reuse_b=*/false);
}

// ---------------- CDNA5 async global->LDS helpers ----------------
__device__ __forceinline__ void async_ld_b32(unsigned lds_byte_addr, const void* g) {
  asm volatile("global_load_async_to_lds_b32 %0, %1, off"
               :: "v"(lds_byte_addr), "v"((unsigned long long)(uintptr_t)g)
               : "memory");
}
__device__ __forceinline__ void wait_async0() {
  asm volatile("s_wait_asynccnt 0x0" ::: "memory");
}
__device__ __forceinline__ unsigned lds_addr_of(const void* p) {
  // generic LDS pointer: addr[31:0] is the LDS byte offset (ISA 10.2 aperture map)
  return (unsigned)(uintptr_t)p;
}

// ---------------- CDNA5 Tensor Data Mover: 2D tile -> LDS ----------------
// rows x cols f32 tile at gsrc (row stride = srcStride elems), contiguous into LDS.
// All descriptor fields are wave-uniform -> SGPR groups via "s" constraints.
__device__ __forceinline__ void tdm_load_tile_f32(unsigned lds_byte_addr, const void* gsrc,
                                                  unsigned rows, unsigned cols,
                                                  unsigned srcStride) {
  unsigned long long ga = (unsigned long long)(uintptr_t)gsrc;
  u32x4 g0;
  g0[0] = 1u;                                   // count=1, is_restore=0, gather off
  g0[1] = lds_byte_addr;                        // D#.lds_addr
  g0[2] = (unsigned)(ga & 0xFFFFFFFFu);         // global_addr[31:0]
  g0[3] = (unsigned)((ga >> 32) & 0x01FFFFFFu)  // global_addr[56:32]
          | (2u << 30);                         // type=2 ("image")
  u32x8 g1;
  g1[0] = 2u << 16;                             // workgroup_mask=0, data_size=2 (4B)
  g1[1] = cols << 16;                           // tensor_dim0[15:0] = cols (tile==tensor)
  g1[2] = rows << 16;                           // tensor_dim0 hi=0, tensor_dim1[15:0]=rows
  g1[3] = cols << 16;                           // tensor_dim1 hi=0, tile_dim0=cols
  g1[4] = rows;                                 // tile_dim1=rows, tile_dim2=0
  g1[5] = srcStride;                            // tensor_dim0_stride[31:0]
  g1[6] = 0u;                                   // stride hi, tensor_dim1_stride lo
  g1[7] = 0u;
  u32x4 g2 = {0u, 0u, 0u, 0u};                  // dims 2/3 unused (zero)
  u32x4 g3 = {0u, 0u, 0u, 0u};
  asm volatile("tensor_load_to_lds %0, %1, %2, %3"
               :: "s"(g0), "s"(g1), "s"(g2), "s"(g3)
               : "memory");
}
__device__ __forceinline__ void wait_tensor0() {
  __builtin_amdgcn_s_wait_tensorcnt(0);         // s_wait_tensorcnt 0
}

#define IMG 2048
#define NIMG 4
#define NTOT (NIMG * IMG * IMG)
#define MEDIAN_RANK 8388607u   // floor(0.5 * (NTOT-1)), quantile method='lower'

// LDS geometry per wave (all f32 words)
#define XS_S 24              // 24x24 patch, contiguous rows (TDM writes dense tile)
#define PS_S 33              // 24x32 product planes (zero-padded), stride 33
#define TS_S 33              // 16x32 intermediate, stride 33
#define XS_SZ (24 * XS_S)    // 576
#define PS_PL (24 * PS_S)    // 792 per channel
#define PS_SZ (3 * PS_PL)    // 2376
#define TS_SZ (16 * TS_S)    // 528
#define WAVE_SZ (XS_SZ + PS_SZ + TS_SZ)  // 3480 words = 13920 B

// ============ Kernel 1: fused Sobel -> products -> separable Gaussian (WMMA) -> R ============
__global__ __launch_bounds__(128) void harris_R_kernel(const float* __restrict__ x,
                                                       float* __restrict__ R) {
  __shared__ __align__(16) float sm[4 * WAVE_SZ];

  const int lane = threadIdx.x & 31;
  const int wave = threadIdx.x >> 5;
  const int tileId = blockIdx.x * 4 + wave;         // 16384 tiles / image
  const int tr = tileId >> 7;                       // tile row  [0,128)
  const int tc = tileId & 127;                      // tile col  [0,128)
  const int b  = blockIdx.y;
  const int r0 = tr * 16, c0 = tc * 16;

  const float* xb = x + (size_t)b * IMG * IMG;
  float* Rb = R + (size_t)b * IMG * IMG;

  float* xs = sm + wave * WAVE_SZ;
  float* ps = xs + XS_SZ;
  float* ts = ps + PS_SZ;

  // ---- load 24x24 x patch (halo 4: sobel 1 + gauss 3) ----
  const bool interior = (tr >= 1) && (tr <= 126) && (tc >= 1) && (tc <= 126);
  if (interior) {
    // Tensor Data Mover: one wave-level DMA of the whole 2D tile into LDS
    const float* src0 = xb + (size_t)(r0 - 4) * IMG + (c0 - 4);
    tdm_load_tile_f32(lds_addr_of(xs), src0, 24u, 24u, (unsigned)IMG);
    wait_tensor0();
  } else {
    for (int i = lane; i < 24 * 24; i += 32) {
      int rr = i / 24, cc = i % 24;
      int gr = r0 - 4 + rr, gc = c0 - 4 + cc;
      float v = 0.0f;
      if ((unsigned)gr < (unsigned)IMG && (unsigned)gc < (unsigned)IMG)
        v = xb[(size_t)gr * IMG + gc];
      xs[rr * XS_S + cc] = v;
    }
  }
  __syncthreads();

  // ---- Sobel + gradient products into 24x32 zero-padded planes (valid 22x22) ----
  for (int i = lane; i < 24 * 32; i += 32) {
    int pr = i >> 5, pc = i & 31;
    float pxx = 0.0f, pyy = 0.0f, pxy = 0.0f;
    if (pr < 22 && pc < 22) {
      const float* c00 = &xs[pr * XS_S + pc];
      float a00 = c00[0], a01 = c00[1], a02 = c00[2];
      float a10 = c00[XS_S], a12 = c00[XS_S + 2];
      float a20 = c00[2 * XS_S], a21 = c00[2 * XS_S + 1], a22 = c00[2 * XS_S + 2];
      float ix = (a02 - a00) + 2.0f * (a12 - a10) + (a22 - a20);
      float iy = (a20 - a00) + 2.0f * (a21 - a01) + (a22 - a02);
      pxx = ix * ix; pyy = iy * iy; pxy = ix * iy;
    }
    ps[0 * PS_PL + pr * PS_S + pc] = pxx;
    ps[1 * PS_PL + pr * PS_S + pc] = pyy;
    ps[2 * PS_PL + pr * PS_S + pc] = pxy;
  }

  // ---- constant banded weight operands (A for vertical, B for horizontal) ----
  const int nn   = lane & 15;                 // N (or M) index within half-wave
  const int aOff = (lane < 16) ? 0 : 2;       // A layout: v0/v1 = K+{0,1} | K+{2,3}
  const int bOff = (lane < 16) ? 0 : 1;       // B layout: v0 = K+{0|1}, v1 = K+{2|3}
  v2f Av[6], Bh[6];
#pragma unroll
  for (int kk = 0; kk < 6; ++kk) {
    int ka = 4 * kk + aOff;
    Av[kk].x = band(nn, ka);
    Av[kk].y = band(nn, ka + 1);
    int kb = 4 * kk + bOff;
    Bh[kk].x = band(nn, kb);
    Bh[kk].y = band(nn, kb + 2);
  }

  v8f S[3];
#pragma unroll
  for (int c = 0; c < 3; ++c) {
    __syncthreads();   // guard ts reuse across channels
    const float* pc_ = ps + c * PS_PL;

    // ---- stage 1: vertical blur  T(16x32) = Gv(16x24) * P(24x32), two N-tiles ----
    v8f t0 = {}, t1 = {};
#pragma unroll
    for (int kk = 0; kk < 6; ++kk) {
      int k0 = 4 * kk + bOff;
      v2f b0, b1;
      b0.x = pc_[k0 * PS_S + nn];
      b0.y = pc_[(k0 + 2) * PS_S + nn];
      b1.x = pc_[k0 * PS_S + 16 + nn];
      b1.y = pc_[(k0 + 2) * PS_S + 16 + nn];
      t0 = wmma4(Av[kk], b0, t0);
      t1 = wmma4(Av[kk], b1, t1);
    }
    // spill T to LDS to convert C/D layout -> A layout
    const int rowM = (lane < 16) ? 0 : 8;
#pragma unroll
    for (int r = 0; r < 8; ++r) {
      ts[(rowM + r) * TS_S + nn]      = t0[r];
      ts[(rowM + r) * TS_S + 16 + nn] = t1[r];
    }
    __syncthreads();

    // ---- stage 2: horizontal blur  S(16x16) = T(16x24) * Gh(24x16) ----
    v8f s = {};
#pragma unroll
    for (int kk = 0; kk < 6; ++kk) {
      int kA = 4 * kk + aOff;
      v2f a;
      a.x = ts[nn * TS_S + kA];
      a.y = ts[nn * TS_S + kA + 1];
      s = wmma4(a, Bh[kk], s);
    }
    S[c] = s;
  }

  // ---- Harris response and store (C/D layout: VGPR r -> M = r or r+8, N = lane%16) ----
  const int rowM = (lane < 16) ? 0 : 8;
#pragma unroll
  for (int r = 0; r < 8; ++r) {
    float sxx = S[0][r], syy = S[1][r], sxy = S[2][r];
    float trc = sxx + syy;
    float Rv = sxx * syy - sxy * sxy - 0.05f * trc * trc;
    Rb[(size_t)(r0 + rowM + r) * IMG + (c0 + nn)] = Rv;
  }
}

// ============ histogram / selection for exact median (monotone float keys) ============
__device__ __forceinline__ unsigned fkey(float f) {
  unsigned u = __float_as_uint(f);
  return (u & 0x80000000u) ? ~u : (u | 0x80000000u);
}

__global__ void zero_kernel(unsigned* __restrict__ p, int n) {
  for (int i = blockIdx.x * blockDim.x + threadIdx.x; i < n; i += gridDim.x * blockDim.x)
    p[i] = 0u;
}

__global__ void hist_hi_kernel(const float4* __restrict__ R4, unsigned* __restrict__ hist,
                               int n4) {
  int stride = gridDim.x * blockDim.x;
  for (int i = blockIdx.x * blockDim.x + threadIdx.x; i < n4; i += stride) {
    if (i + stride < n4) __builtin_prefetch(&R4[i + stride], 0, 1);  // global_prefetch_b8
    float4 v = R4[i];
    atomicAdd(&hist[fkey(v.x) >> 16], 1u);
    atomicAdd(&hist[fkey(v.y) >> 16], 1u);
    atomicAdd(&hist[fkey(v.z) >> 16], 1u);
    atomicAdd(&hist[fkey(v.w) >> 16], 1u);
  }
}

__global__ void hist_lo_kernel(const float4* __restrict__ R4, unsigned* __restrict__ hist,
                               const unsigned* __restrict__ state, int n4) {
  unsigned hi = state[0];
  int stride = gridDim.x * blockDim.x;
  for (int i = blockIdx.x * blockDim.x + threadIdx.x; i < n4; i += stride) {
    if (i + stride < n4) __builtin_prefetch(&R4[i + stride], 0, 1);
    float4 v = R4[i];
    unsigned k0 = fkey(v.x), k1 = fkey(v.y), k2 = fkey(v.z), k3 = fkey(v.w);
    if ((k0 >> 16) == hi) atomicAdd(&hist[k0 & 0xFFFFu], 1u);
    if ((k1 >> 16) == hi) atomicAdd(&hist[k1 & 0xFFFFu], 1u);
    if ((k2 >> 16) == hi) atomicAdd(&hist[k2 & 0xFFFFu], 1u);
    if ((k3 >> 16) == hi) atomicAdd(&hist[k3 & 0xFFFFu], 1u);
  }
}

__global__ void select_kernel(const unsigned* __restrict__ hist, unsigned* state, int mode) {
  __shared__ unsigned partial[256];
  int t = threadIdx.x;
  unsigned s = 0;
  for (int j = 0; j < 256; ++j) s += hist[t * 256 + j];
  partial[t] = s;
  __syncthreads();
  if (t == 0) {
    unsigned target = (mode == 0) ? MEDIAN_RANK : state[1];
    unsigned cum = 0;
    int chunk = 0;
    while (chunk < 255 && cum + partial[chunk] <= target) { cum += partial[chunk]; ++chunk; }
    int bin = chunk * 256, binEnd = chunk * 256 + 255;
    while (bin < binEnd && cum + hist[bin] <= target) { cum += hist[bin]; ++bin; }
    if (mode == 0) {
      state[0] = (unsigned)bin;          // selected high-16 bucket
      state[1] = target - cum;           // remaining rank inside bucket
    } else {
      unsigned key = (state[0] << 16) | (unsigned)bin;
      unsigned fb = (key & 0x80000000u) ? (key & 0x7FFFFFFFu) : ~key;  // un-map
      state[2] = fb;                     // median float bits
    }
  }
}

// ============ final: threshold + 7x7 maxpool + NMS mask * R ============
__global__ __launch_bounds__(256) void finalize_kernel(const float* __restrict__ R,
                                                       const unsigned* __restrict__ state,
                                                       float* __restrict__ out) {
  __shared__ __align__(16) float rth[38 * 40];   // thresholded (OOB = -inf)
  __shared__ __align__(16) float rraw[38 * 40];  // raw R
  float med = __uint_as_float(state[2]);

  int b = blockIdx.y;
  int ty = blockIdx.x >> 6, tx = blockIdx.x & 63;   // 64x64 tiles of 32x32
  int r0 = ty * 32, c0 = tx * 32;
  const float* Rb = R + (size_t)b * IMG * IMG;

  const bool interior = (ty >= 1) && (ty <= 62) && (tx >= 1) && (tx <= 62);
  if (interior) {
    // async DMA halo tile straight into LDS (b32: -3 halo breaks 16B alignment)
    const float* src0 = Rb + (size_t)(r0 - 3) * IMG + (c0 - 3);
    const unsigned ldsb = lds_addr_of(rraw);
    for (int i = threadIdx.x; i < 38 * 38; i += blockDim.x) {
      int rr = i / 38, cc = i % 38;
      async_ld_b32(ldsb + (unsigned)(rr * 40 + cc) * 4u, src0 + (size_t)rr * IMG + cc);
    }
    wait_async0();
    __syncthreads();
    for (int j = threadIdx.x; j < 38 * 40; j += blockDim.x) {
      float raw = rraw[j];
      rth[j] = (raw < med) ? 0.0f : raw;
    }
  } else {
    for (int i = threadIdx.x; i < 38 * 38; i += blockDim.x) {
      int rr = i / 38, cc = i % 38;
      int gr = r0 - 3 + rr, gc = c0 - 3 + cc;
      float raw = 0.0f, th = -INFINITY;
      if ((unsigned)gr < (unsigned)IMG && (unsigned)gc < (unsigned)IMG) {
        raw = Rb[(size_t)gr * IMG + gc];
        th = (raw < med) ? 0.0f : raw;
      }
      rth[rr * 40 + cc] = th;
      rraw[rr * 40 + cc] = raw;
    }
  }
  __syncthreads();

  for (int i = threadIdx.x; i < 32 * 32; i += blockDim.x) {
    int rr = i >> 5, cc = i & 31;
    float m = -INFINITY;
#pragma unroll
    for (int dy = 0; dy < 7; ++dy)
#pragma unroll
      for (int dx = 0; dx < 7; ++dx)
        m = fmaxf(m, rth[(rr + dy) * 40 + (cc + dx)]);
    float thc = rth[(rr + 3) * 40 + (cc + 3)];
    float rc  = rraw[(rr + 3) * 40 + (cc + 3)];
    out[(size_t)b * IMG * IMG + (size_t)(r0 + rr) * IMG + (c0 + cc)] =
        (thc == m) ? rc : 0.0f;
  }
}

// ============ host launcher ============
extern "C" void kernel_launch(void* const* d_in, const int* in_sizes, int n_in,
                              void* d_out, int out_size, void* d_ws, size_t ws_size,
                              hipStream_t stream) {
  const float* x = (const float*)d_in[0];   // (4,1,2048,2048) f32
  float* out = (float*)d_out;               // (4,1,2048,2048) f32

  // workspace layout
  float* R = (float*)d_ws;                                   // 67,108,864 B
  unsigned* hist0 = (unsigned*)((char*)d_ws + (size_t)NTOT * 4);
  unsigned* hist1 = hist0 + 65536;
  unsigned* state = hist1 + 65536;                           // [hi, remRank, medianBits]

  zero_kernel<<<256, 256, 0, stream>>>(hist0, 2 * 65536);

  // fused Sobel -> products -> separable Gaussian (WMMA) -> R
  harris_R_kernel<<<dim3(4096, NIMG), 128, 0, stream>>>(x, R);

  // exact median via 2-level 16-bit radix select
  hist_hi_kernel<<<4096, 256, 0, stream>>>((const float4*)R, hist0, NTOT / 4);
  select_kernel<<<1, 256, 0, stream>>>(hist0, state, 0);
  hist_lo_kernel<<<4096, 256, 0, stream>>>((const float4*)R, hist1, state, NTOT / 4);
  select_kernel<<<1, 256, 0, stream>>>(hist1, state, 1);

  // threshold + 7x7 maxpool + NMS * R
  finalize_kernel<<<dim3(4096, NIMG), 256, 0, stream>>>(R, state, out);
}